// Yolov1Loss_58334245814915
// MI455X (gfx1250) — compile-verified
//
#include <hip/hip_runtime.h>

typedef __attribute__((ext_vector_type(2))) float v2f;
typedef __attribute__((ext_vector_type(8))) float v8f;

#define CPB 256   // cells per block
#define CF  30    // floats per cell (YOLOv1: 2 boxes*5 + 20 classes)

// Cross-block reduction finished with V_WMMA_F32_16X16X4_F32:
// A(16x4 f32) holds the 8 per-wave partials (lanes 0-15 -> K=0 slot,
// lanes 16-31 -> K=2 slot, other K slots zero). B(4x16) = all ones, so
// D[m][n] = partial[m] + partial[m+16] replicated across columns.
// Summing the 8 accumulator VGPRs in-lane gives rows 0-7 (lanes 0-15)
// and rows 8-15 (lanes 16-31); one xor-16 shuffle adds the halves.
__device__ __forceinline__ float block_reduce_wmma(float v, float* s_red) {
    const int tid = threadIdx.x;
#pragma unroll
    for (int m = 16; m >= 1; m >>= 1) v += __shfl_xor(v, m, 32);
    if ((tid & 31) == 0) s_red[tid >> 5] = v;
    __syncthreads();
    float total = 0.0f;
    if (tid < 32) {   // whole wave 0 active -> EXEC all ones for WMMA
        const int nwaves = (int)(blockDim.x >> 5);
        float p = (tid < nwaves) ? s_red[tid] : 0.0f;
        v2f a; a[0] = p;    a[1] = 0.0f;
        v2f b; b[0] = 1.0f; b[1] = 1.0f;
        v8f c = {};
        v8f d = __builtin_amdgcn_wmma_f32_16x16x4_f32(
            /*neg_a=*/false, a, /*neg_b=*/false, b,
            /*c_mod=*/(short)0, c, /*reuse_a=*/false, /*reuse_b=*/false);
        float s = d[0] + d[1] + d[2] + d[3] + d[4] + d[5] + d[6] + d[7];
        s += __shfl_xor(s, 16, 32);
        total = s;
    }
    return total;  // valid for tid < 32 (in particular tid == 0)
}

__global__ void __launch_bounds__(CPB)
yolo_partial_kernel(const float* __restrict__ inp,
                    const float* __restrict__ tgt,
                    float* __restrict__ partials,
                    int n_cells) {
    __shared__ float s_in[CPB * CF];
    __shared__ float s_tg[CPB * CF];
    __shared__ float s_red[CPB / 32];

    const int tid = threadIdx.x;
    const long long base = (long long)blockIdx.x * (CPB * CF);
    const long long totf = (long long)n_cells * CF;
    long long rem = totf - base;
    const int nf  = (rem < (long long)(CPB * CF)) ? (int)rem : (CPB * CF);
    const int nv4 = nf >> 2;

    // Stage 256 cells of both tensors into LDS with coalesced b128 loads.
    const float4* gin = reinterpret_cast<const float4*>(inp + base);
    const float4* gtg = reinterpret_cast<const float4*>(tgt + base);
    float4* lin = reinterpret_cast<float4*>(s_in);
    float4* ltg = reinterpret_cast<float4*>(s_tg);
    for (int i = tid; i < nv4; i += CPB) { lin[i] = gin[i]; ltg[i] = gtg[i]; }
    for (int i = (nv4 << 2) + tid; i < nf; i += CPB) {
        s_in[i] = inp[base + i]; s_tg[i] = tgt[base + i];
    }
    __syncthreads();

    float loss = 0.0f;
    const long long cell = (long long)blockIdx.x * CPB + tid;
    if (cell < (long long)n_cells) {
        const float* in = s_in + tid * CF;
        const float* tg = s_tg + tid * CF;
        const float gs = 7.0f;

        const float cm = (tg[4] > 0.0f) ? 1.0f : 0.0f;
        const float nm = 1.0f - cm;

        // no-object confidence loss (LAMBDA_NOOBJ = 0.5 folded in)
        const float d0 = in[4] - tg[4];
        const float d1 = in[9] - tg[9];
        loss += 0.5f * nm * (d0 * d0 + d1 * d1);

        // target box 0 extents (match reference: areas via max-min products)
        const float tcx = tg[0] / gs, tcy = tg[1] / gs;
        const float tminx = tcx - 0.5f * tg[2], tmaxx = tcx + 0.5f * tg[2];
        const float tminy = tcy - 0.5f * tg[3], tmaxy = tcy + 0.5f * tg[3];
        const float area_t = (tmaxx - tminx) * (tmaxy - tminy);

        float iou[2];
#pragma unroll
        for (int b = 0; b < 2; ++b) {
            const float* p = in + 5 * b;
            const float pcx = p[0] / gs, pcy = p[1] / gs;
            const float pminx = pcx - 0.5f * p[2], pmaxx = pcx + 0.5f * p[2];
            const float pminy = pcy - 0.5f * p[3], pmaxy = pcy + 0.5f * p[3];
            float iw = fminf(pmaxx, tmaxx) - fmaxf(pminx, tminx);
            float ih = fminf(pmaxy, tmaxy) - fmaxf(pminy, tminy);
            iw = fmaxf(iw, 0.0f); ih = fmaxf(ih, 0.0f);
            const float inter  = iw * ih;
            const float area_p = (pmaxx - pminx) * (pmaxy - pminy);
            iou[b] = inter / (area_p + area_t - inter);
        }
        const float max_iou = fmaxf(iou[0], iou[1]);
        const float r0 = (iou[0] >= iou[1]) ? 1.0f : 0.0f;  // argmax tie -> box 0

#pragma unroll
        for (int b = 0; b < 2; ++b) {
            const float w = cm * ((b == 0) ? r0 : 1.0f - r0);
            const float* p = in + 5 * b;
            const float* q = tg + 5 * b;
            const float ex = p[0] - q[0], ey = p[1] - q[1];
            loss += 0.5f * w * (ex * ex + ey * ey);               // LAMBDA_COORD
            const float sw = sqrtf(p[2]) - sqrtf(q[2]);
            const float sh = sqrtf(p[3]) - sqrtf(q[3]);
            loss += 0.5f * w * (sw * sw + sh * sh);               // LAMBDA_COORD
            const float eo = p[4] - max_iou;
            loss += w * eo * eo;
        }

        float cls = 0.0f;
#pragma unroll
        for (int k = 0; k < 20; ++k) {
            const float e = in[10 + k] - tg[10 + k];
            cls += e * e;
        }
        loss += cm * cls;
    }

    const float tot = block_reduce_wmma(loss, s_red);
    if (tid == 0) partials[blockIdx.x] = tot;
}

__global__ void __launch_bounds__(CPB)
yolo_final_kernel(const float* __restrict__ partials,
                  float* __restrict__ out, int n, float scale) {
    __shared__ float s_red[CPB / 32];
    const int tid = threadIdx.x;
    float v = 0.0f;
    for (int i = tid; i < n; i += CPB) v += partials[i];  // fixed order: deterministic
    const float tot = block_reduce_wmma(v, s_red);
    if (tid == 0) out[0] = tot * scale;
}

extern "C" void kernel_launch(void* const* d_in, const int* in_sizes, int n_in,
                              void* d_out, int out_size, void* d_ws, size_t ws_size,
                              hipStream_t stream) {
    (void)n_in; (void)out_size; (void)ws_size;
    const float* inp = (const float*)d_in[0];
    const float* tgt = (const float*)d_in[1];
    float* out      = (float*)d_out;
    float* partials = (float*)d_ws;

    const long long total = (long long)in_sizes[0];      // B*7*7*30
    const int n_cells = (int)(total / CF);               // B*49
    const int blocks  = (n_cells + CPB - 1) / CPB;
    const int bs      = n_cells / 49;                    // batch size

    yolo_partial_kernel<<<blocks, CPB, 0, stream>>>(inp, tgt, partials, n_cells);
    yolo_final_kernel<<<1, CPB, 0, stream>>>(partials, out, blocks, 1.0f / (float)bs);
}